// RecyclingEmbedder_49727131353931
// MI455X (gfx1250) — compile-verified
//
#include <hip/hip_runtime.h>
#include <stdint.h>

#define DIMC 48          // channel dim
#define NBINS 38
#define PTPB 192         // pair-kernel threads per block: 6 wave32, divisible by 12 -> affine DMA addressing
#define STPB 256         // single-kernel threads per block
#define ROWSTRIDE 52     // padded floats per LDS row: 208B, 16B-aligned, low bank conflict
#define F4_PER_ROW 12    // 48 floats = 12 float4
#define LDS_ROW_B (ROWSTRIDE * 4)            // 208 bytes
#define K_LDS_STEP (16 * LDS_ROW_B)          // per-k LDS advance: 16 rows = 3328 B
#define K_GLB_STEP (PTPB * 16)               // per-k global advance: 192 float4 = 3072 B

// ---------- CDNA5 (gfx1250) async global<->LDS DMA helpers ----------
// Non-temporal hints: the 2x192MB pair stream is touched once; keep it from
// thrashing the 192MB L2 so W/x/gamma/beta stay hot.
__device__ __forceinline__ void async_load_f4_nt(void* lds_ptr, const void* gptr) {
  auto l = (__attribute__((address_space(3))) char*)lds_ptr;   // raw LDS byte offset in a VGPR
  asm volatile("global_load_async_to_lds_b128 %0, %1, off th:TH_LOAD_NT"
               :: "v"(l), "v"(gptr)
               : "memory");
}
__device__ __forceinline__ void async_store_f4_nt(const void* gptr, void* lds_ptr) {
  auto l = (__attribute__((address_space(3))) char*)lds_ptr;
  asm volatile("global_store_async_from_lds_b128 %0, %1, off th:TH_STORE_NT"
               :: "v"(gptr), "v"(l)
               : "memory");
}
__device__ __forceinline__ void wait_asynccnt0() {
  asm volatile("s_wait_asynccnt 0" ::: "memory");
}

// ---------- pair: LN(pair) + W[bin(d_ij)] + b ----------
__global__ __launch_bounds__(PTPB) void pair_ln_emb_kernel(
    const float* __restrict__ x,       // [L,3]
    const float* __restrict__ pairin,  // [L,L,48]
    const float* __restrict__ W,       // [38,48]
    const float* __restrict__ bvec,    // [48]
    const float* __restrict__ gamma,   // [48]
    const float* __restrict__ beta,    // [48]
    float* __restrict__ pairout,       // [L,L,48]
    int L, long long numPairs)
{
  __shared__ float smem[PTPB * ROWSTRIDE];   // 39,936 B -> 8 blocks / 320KB WGP
  const int tid = threadIdx.x;
  const long long chunkBase = (long long)blockIdx.x * PTPB;  // first pair of this chunk
  const bool fullBlock = (chunkBase + PTPB) <= numPairs;     // UNIFORM: scalar branch only

  // Affine chunk mapping (PTPB % 12 == 0): g = tid + k*PTPB
  //   p = p0 + 16k,  c = c0   with  p0 = tid/12, c0 = tid%12  (single /12)
  const int p0 = tid / F4_PER_ROW;
  const int c0 = tid - p0 * F4_PER_ROW;
  char* lbase = (char*)smem + p0 * LDS_ROW_B + c0 * 16;

  // ---- Phase 1: coalesced async DMA global -> LDS (padded rows) ----
  {
    const char* gbase = (const char*)(pairin + chunkBase * DIMC) + (size_t)tid * 16;
    if (fullBlock) {
      #pragma unroll
      for (int k = 0; k < F4_PER_ROW; ++k)
        async_load_f4_nt(lbase + k * K_LDS_STEP, gbase + (size_t)k * K_GLB_STEP);
    } else {
      #pragma unroll
      for (int k = 0; k < F4_PER_ROW; ++k)
        if (chunkBase + p0 + 16 * k < numPairs)
          async_load_f4_nt(lbase + k * K_LDS_STEP, gbase + (size_t)k * K_GLB_STEP);
    }
  }
  wait_asynccnt0();
  __syncthreads();

  // ---- Phase 2: one pair-row per lane ----
  long long P  = chunkBase + tid;
  long long Pc = (P < numPairs) ? P : (numPairs - 1);

  float4 v[F4_PER_ROW];
  #pragma unroll
  for (int c = 0; c < F4_PER_ROW; ++c)
    v[c] = *(const float4*)&smem[tid * ROWSTRIDE + c * 4];

  float sum = 0.f, ssq = 0.f;
  #pragma unroll
  for (int c = 0; c < F4_PER_ROW; ++c) {
    sum += v[c].x + v[c].y + v[c].z + v[c].w;
    ssq += v[c].x * v[c].x + v[c].y * v[c].y + v[c].z * v[c].z + v[c].w * v[c].w;
  }
  const float inv = 1.0f / (float)DIMC;
  float mu   = sum * inv;
  float var  = fmaxf(ssq * inv - mu * mu, 0.0f);
  float rstd = rsqrtf(var + 1e-5f);

  // distance -> nearest bin (uniform linspace => closed-form argmin)
  int i = (int)(Pc / L);
  int j = (int)(Pc - (long long)i * L);
  float dx = x[3 * i + 0] - x[3 * j + 0];
  float dy = x[3 * i + 1] - x[3 * j + 1];
  float dz = x[3 * i + 2] - x[3 * j + 2];
  float d  = sqrtf(dx * dx + dy * dy + dz * dz + 1e-12f);
  const float bin0 = 3.25f;
  const float invStep = 37.0f / 17.5f;     // 1 / ((20.75-3.25)/37)
  int idx = (int)floorf((d - bin0) * invStep + 0.5f);
  idx = idx < 0 ? 0 : (idx > NBINS - 1 ? NBINS - 1 : idx);

  const float4* Wrow = (const float4*)(W + (size_t)idx * DIMC);  // 7KB table: cache-resident
  const float4* g4p  = (const float4*)gamma;
  const float4* be4p = (const float4*)beta;
  const float4* bb4p = (const float4*)bvec;

  #pragma unroll
  for (int c = 0; c < F4_PER_ROW; ++c) {
    float4 w4 = Wrow[c], g4 = g4p[c], be4 = be4p[c], bb4 = bb4p[c];
    float4 o;
    o.x = (v[c].x - mu) * rstd * g4.x + be4.x + w4.x + bb4.x;
    o.y = (v[c].y - mu) * rstd * g4.y + be4.y + w4.y + bb4.y;
    o.z = (v[c].z - mu) * rstd * g4.z + be4.z + w4.z + bb4.z;
    o.w = (v[c].w - mu) * rstd * g4.w + be4.w + w4.w + bb4.w;
    *(float4*)&smem[tid * ROWSTRIDE + c * 4] = o;
  }
  __syncthreads();   // LDS writes visible before the async engine reads them

  // ---- Phase 3: coalesced async DMA LDS -> global (s_endpgm drains ASYNCcnt) ----
  {
    char* gbase = (char*)(pairout + chunkBase * DIMC) + (size_t)tid * 16;
    if (fullBlock) {
      #pragma unroll
      for (int k = 0; k < F4_PER_ROW; ++k)
        async_store_f4_nt(gbase + (size_t)k * K_GLB_STEP, lbase + k * K_LDS_STEP);
    } else {
      #pragma unroll
      for (int k = 0; k < F4_PER_ROW; ++k)
        if (chunkBase + p0 + 16 * k < numPairs)
          async_store_f4_nt(gbase + (size_t)k * K_GLB_STEP, lbase + k * K_LDS_STEP);
    }
  }
}

// ---------- single: LN(single), tiny ----------
__global__ __launch_bounds__(STPB) void single_ln_kernel(
    const float* __restrict__ sin_, const float* __restrict__ gamma,
    const float* __restrict__ beta, float* __restrict__ out, int nRows)
{
  int r = blockIdx.x * STPB + threadIdx.x;
  if (r >= nRows) return;
  const float4* src = (const float4*)(sin_ + (size_t)r * DIMC);
  float4 v[F4_PER_ROW];
  float sum = 0.f, ssq = 0.f;
  #pragma unroll
  for (int c = 0; c < F4_PER_ROW; ++c) {
    v[c] = src[c];
    sum += v[c].x + v[c].y + v[c].z + v[c].w;
    ssq += v[c].x * v[c].x + v[c].y * v[c].y + v[c].z * v[c].z + v[c].w * v[c].w;
  }
  const float inv = 1.0f / (float)DIMC;
  float mu   = sum * inv;
  float var  = fmaxf(ssq * inv - mu * mu, 0.0f);
  float rstd = rsqrtf(var + 1e-5f);
  const float4* g4p  = (const float4*)gamma;
  const float4* be4p = (const float4*)beta;
  float4* dst = (float4*)(out + (size_t)r * DIMC);
  #pragma unroll
  for (int c = 0; c < F4_PER_ROW; ++c) {
    float4 g4 = g4p[c], be4 = be4p[c], o;
    o.x = (v[c].x - mu) * rstd * g4.x + be4.x;
    o.y = (v[c].y - mu) * rstd * g4.y + be4.y;
    o.z = (v[c].z - mu) * rstd * g4.z + be4.z;
    o.w = (v[c].w - mu) * rstd * g4.w + be4.w;
    dst[c] = o;
  }
}

extern "C" void kernel_launch(void* const* d_in, const int* in_sizes, int n_in,
                              void* d_out, int out_size, void* d_ws, size_t ws_size,
                              hipStream_t stream) {
  (void)n_in; (void)out_size; (void)d_ws; (void)ws_size;
  // setup_inputs order: x, single, pair, W, b, gamma_pair, beta_pair, gamma_msa, beta_msa
  const float* x      = (const float*)d_in[0];
  const float* single = (const float*)d_in[1];
  const float* pair   = (const float*)d_in[2];
  const float* W      = (const float*)d_in[3];
  const float* b      = (const float*)d_in[4];
  const float* gpair  = (const float*)d_in[5];
  const float* bpair  = (const float*)d_in[6];
  const float* gmsa   = (const float*)d_in[7];
  const float* bmsa   = (const float*)d_in[8];
  float* out = (float*)d_out;

  const int L = in_sizes[0] / 3;                 // x is [L,3]
  const long long numPairs = (long long)L * (long long)L;
  float* single_out = out;                       // outputs concatenated: single_out, pair_out
  float* pair_out   = out + (size_t)L * DIMC;

  int sblocks = (L + STPB - 1) / STPB;
  single_ln_kernel<<<sblocks, STPB, 0, stream>>>(single, gmsa, bmsa, single_out, L);

  long long pblocks = (numPairs + PTPB - 1) / PTPB;
  pair_ln_emb_kernel<<<(unsigned)pblocks, PTPB, 0, stream>>>(
      x, pair, W, b, gpair, bpair, pair_out, L, numPairs);
}